// MemoryGraph_35347580846916
// MI455X (gfx1250) — compile-verified
//
#include <hip/hip_runtime.h>
#include <math.h>

#define NN   4096
#define KCON 64
#define DD   64
#define CMEM 8
#define HIDN 32
#define NB   8
#define BSZ  8
#define NG   2
#define BPG  4
#define BB   4
#define TT   64
#define NUPD 16

typedef float v2f __attribute__((ext_vector_type(2)));
typedef float v8f __attribute__((ext_vector_type(8)));
typedef int   v4i __attribute__((ext_vector_type(4)));

#if __has_builtin(__builtin_amdgcn_global_load_async_to_lds_b128) && \
    __has_builtin(__builtin_amdgcn_s_wait_asynccnt)
#define HAS_ASYNC_LDS 1
#else
#define HAS_ASYNC_LDS 0
#endif

__device__ __forceinline__ float sigm(float x) { return 1.0f / (1.0f + expf(-x)); }

// ---------------------------------------------------------------------------
// Init: copy h / prev_messages into workspace state; build sigmoid(ro) padded
// to 16 rows (rows 8..15 = 0) so the WMMA readout A-operand loads are
// unconditional (no exec-mask juggling in the hot loop).
// ---------------------------------------------------------------------------
__global__ __launch_bounds__(256) void k_init(
    const float* __restrict__ hin, const float* __restrict__ pmin,
    const float* __restrict__ row,
    float* __restrict__ ws_h, float* __restrict__ ws_pm, float* __restrict__ rosig)
{
  const int i = blockIdx.x * 256 + threadIdx.x;
  const int NH = BB * NN * DD;
  if (i < NH) { ws_h[i] = hin[i]; ws_pm[i] = pmin[i]; }
  if (i < 16 * NN) rosig[i] = (i < CMEM * NN) ? sigm(row[i]) : 0.0f;
}

// ---------------------------------------------------------------------------
// Modulator: per-neuron MLP -> eff_prim / eff_key / eff_decay.
// One wave per neuron (lane = hidden unit), streams fc1_w (167MB) coalesced.
// ---------------------------------------------------------------------------
__global__ __launch_bounds__(256) void k_modulator(
    const float* __restrict__ hin, const float* __restrict__ tp,
    const float* __restrict__ tk, const float* __restrict__ prim,
    const float* __restrict__ keyp, const float* __restrict__ dlog,
    const float* __restrict__ fc1w, const float* __restrict__ fc1b,
    const float* __restrict__ fc2w, const float* __restrict__ fc2b,
    const float* __restrict__ mlr_logit,
    float* __restrict__ eff_prim, float* __restrict__ eff_key,
    float* __restrict__ eff_dec)
{
  __shared__ float smod[8][5][4][DD];   // 40 KB
  __shared__ float sx[8][4][HIDN];      // 4 KB
  __shared__ float s3[8][4][3];
  __shared__ float snrm[8][2][4];
  const int w  = threadIdx.x >> 5;
  const int l  = threadIdx.x & 31;
  const int n  = blockIdx.x * 8 + w;
  const int d0 = l << 1;

  { // stage mod_in segments (lane covers d = 2l, 2l+1)
    const int base = n * DD + d0;
#pragma unroll
    for (int b = 0; b < BB; ++b) {
      float2 v0 = *(const float2*)(hin + b * NN * DD + base);
      smod[w][0][b][d0] = v0.x; smod[w][0][b][d0 + 1] = v0.y;
      float2 v1 = *(const float2*)(tp + b * NN * DD + base);
      smod[w][1][b][d0] = v1.x; smod[w][1][b][d0 + 1] = v1.y;
      float2 v2 = *(const float2*)(tk + b * NN * DD + base);
      smod[w][2][b][d0] = v2.x; smod[w][2][b][d0 + 1] = v2.y;
    }
    float2 vp = *(const float2*)(prim + base);
    smod[w][3][0][d0] = vp.x; smod[w][3][0][d0 + 1] = vp.y;
    float2 vk = *(const float2*)(keyp + base);
    smod[w][4][0][d0] = vk.x; smod[w][4][0][d0 + 1] = vk.y;
  }
  __syncthreads();

  float acc0, acc1, acc2, acc3;
  acc0 = acc1 = acc2 = acc3 = fc1b[n * HIDN + l];
  const float* wp = fc1w + (size_t)n * (5 * DD * HIDN) + l;
  for (int s = 0; s < 3; ++s) {              // batch-dependent segments
    const float* wr = wp + s * DD * HIDN;
#pragma unroll 4
    for (int d = 0; d < DD; ++d) {
      const float wv = wr[d * HIDN];
      acc0 = fmaf(wv, smod[w][s][0][d], acc0);
      acc1 = fmaf(wv, smod[w][s][1][d], acc1);
      acc2 = fmaf(wv, smod[w][s][2][d], acc2);
      acc3 = fmaf(wv, smod[w][s][3][d], acc3);
    }
  }
  for (int s = 3; s < 5; ++s) {              // batch-independent segments
    const float* wr = wp + s * DD * HIDN;
#pragma unroll 4
    for (int d = 0; d < DD; ++d) {
      const float wv = wr[d * HIDN];
      const float m = smod[w][s][0][d];
      acc0 = fmaf(wv, m, acc0); acc1 = fmaf(wv, m, acc1);
      acc2 = fmaf(wv, m, acc2); acc3 = fmaf(wv, m, acc3);
    }
  }
  sx[w][0][l] = tanhf(acc0); sx[w][1][l] = tanhf(acc1);
  sx[w][2][l] = tanhf(acc2); sx[w][3][l] = tanhf(acc3);
  __syncthreads();

  if (l < 12) {                               // fc2: 4 batches x 3 outputs
    const int b = l & 3, o = l >> 2;
    float s = fc2b[n * 3 + o];
#pragma unroll 4
    for (int hh = 0; hh < HIDN; ++hh)
      s = fmaf(sx[w][b][hh], fc2w[n * 96 + hh * 3 + o], s);
    s3[w][b][o] = s;
  }
  if (l < 8) {                                // trace norms
    const int which = l >> 2, b = l & 3;
    float ss = 0.f;
    for (int d = 0; d < DD; ++d) {
      const float v = smod[w][1 + which][b][d];
      ss = fmaf(v, v, ss);
    }
    snrm[w][which][b] = fmaxf(sqrtf(ss), 1e-8f);
  }
  __syncthreads();

  const float mlr = sigm(mlr_logit[0]);
  if (l < 4) eff_dec[l * NN + n] = sigm(dlog[n] + s3[w][l][2]);

  const float p0 = smod[w][3][0][d0], p1 = smod[w][3][0][d0 + 1];
  const float k0 = smod[w][4][0][d0], k1 = smod[w][4][0][d0 + 1];
#pragma unroll
  for (int b = 0; b < BB; ++b) {
    const float gp  = mlr * tanhf(s3[w][b][0]);
    const float gk  = mlr * tanhf(s3[w][b][1]);
    const float itn = 1.0f / snrm[w][0][b];
    const float ikn = 1.0f / snrm[w][1][b];
    float2 ep, ek;
    ep.x = fmaf(gp, smod[w][1][b][d0] * itn, p0);
    ep.y = fmaf(gp, smod[w][1][b][d0 + 1] * itn, p1);
    ek.x = fmaf(gk, smod[w][2][b][d0] * ikn, k0);
    ek.y = fmaf(gk, smod[w][2][b][d0 + 1] * ikn, k1);
    *(float2*)(eff_prim + (b * NN + n) * DD + d0) = ep;
    *(float2*)(eff_key  + (b * NN + n) * DD + d0) = ek;
  }
}

// ---------------------------------------------------------------------------
// Graph update at t (t % 4 == 0). Block = 64 thr (2 waves), wave = (b,n);
// both waves share one neuron n. dendrite_branch_w slab (16KB) is DMA'd
// into LDS with GLOBAL_LOAD_ASYNC_TO_LDS_B128 at kernel entry, overlapping
// the neighbor gather + routing-dot pass; waited with s_wait_asynccnt.
// ---------------------------------------------------------------------------
__global__ __launch_bounds__(64) void k_update(
    const float* __restrict__ pm_in, float* __restrict__ pm_out,
    float* __restrict__ hbuf,
    const float* __restrict__ eff_prim, const float* __restrict__ eff_key,
    const float* __restrict__ eff_dec,
    const float* __restrict__ dbw, const float* __restrict__ dgw,
    const float* __restrict__ injw, const float* __restrict__ cc,
    const int* __restrict__ conn, const int t)
{
  __shared__ __align__(16) float snm[2][KCON][68];  // 34 KB, pad 4 -> 16B rows
  __shared__ __align__(16) float swbr[NB * BSZ * DD]; // 16 KB branch weights
  __shared__ __align__(16) float sek[2][DD];
  __shared__ float srt[2][KCON];
  const int w  = threadIdx.x >> 5;
  const int l  = threadIdx.x & 31;
  const int n  = blockIdx.x >> 1;
  const int b  = ((blockIdx.x & 1) << 1) + w;
  const int d0 = l << 1;

#if HAS_ASYNC_LDS
  { // async DMA: 16KB slab, 64 threads x 16 x b128
    const float* src = dbw + (size_t)n * (NB * BSZ * DD);
#pragma unroll
    for (int i = 0; i < 16; ++i) {
      const int off = (threadIdx.x + i * 64) * 4;      // float index
      v4i* g = (v4i*)(src + off);
      v4i* s = (v4i*)(&swbr[off]);
      __builtin_amdgcn_global_load_async_to_lds_b128(g, s, 0, 0);
    }
  }
#endif

  { float2 v = *(const float2*)(eff_key + (b * NN + n) * DD + d0);
    sek[w][d0] = v.x; sek[w][d0 + 1] = v.y; }
  const int ci0 = conn[n * KCON + l];
  const int ci1 = conn[n * KCON + l + 32];
  __builtin_prefetch(dgw + (size_t)n * (NG * BPG * DD), 0, 0);
  __syncthreads();

  // Pass 1: each lane streams 2 neighbor rows; dot with eff_key + stash in LDS.
  const float* r0 = pm_in + (b * NN + ci0) * DD;
  const float* r1 = pm_in + (b * NN + ci1) * DD;
  float dot0 = 0.f, dot1 = 0.f;
#pragma unroll 4
  for (int c = 0; c < 16; ++c) {
    const float4 e  = *(const float4*)&sek[w][c * 4];
    const float4 v0 = *(const float4*)(r0 + c * 4);
    dot0 += v0.x * e.x + v0.y * e.y + v0.z * e.z + v0.w * e.w;
    *(float4*)&snm[w][l][c * 4] = v0;
    const float4 v1 = *(const float4*)(r1 + c * 4);
    dot1 += v1.x * e.x + v1.y * e.y + v1.z * e.z + v1.w * e.w;
    *(float4*)&snm[w][l + 32][c * 4] = v1;
  }
  srt[w][l]      = sigm(dot0);
  srt[w][l + 32] = sigm(dot1);
#if HAS_ASYNC_LDS
  __builtin_amdgcn_s_wait_asynccnt(0);  // own wave's DMA done (barrier next)
#endif
  __syncthreads();

#if HAS_ASYNC_LDS
  const float* wbase = swbr;                           // LDS-staged weights
#else
  const float* wbase = dbw + (size_t)n * (NB * BSZ * DD);
#endif

  // Pass 2: lane covers d = 2l, 2l+1.
  float br0[NB], br1[NB];
#pragma unroll
  for (int v = 0; v < NB; ++v) {
    float a0 = 0.f, a1 = 0.f;
#pragma unroll
    for (int s = 0; s < BSZ; ++s) {
      const int   k  = v * BSZ + s;
      const float r  = srt[w][k];
      const float2 nm = *(const float2*)&snm[w][k][d0];
      const float2 wv = *(const float2*)(wbase + (v * BSZ + s) * DD + d0);
      a0 = fmaf(wv.x * r, nm.x, a0);
      a1 = fmaf(wv.y * r, nm.y, a1);
    }
    br0[v] = tanhf(a0); br1[v] = tanhf(a1);
  }
  float rcv0 = 0.f, rcv1 = 0.f;
#pragma unroll
  for (int g = 0; g < NG; ++g) {
    float a0 = 0.f, a1 = 0.f;
#pragma unroll
    for (int p = 0; p < BPG; ++p) {
      const float2 wv = *(const float2*)(dgw + (((size_t)n * NG + g) * BPG + p) * DD + d0);
      a0 = fmaf(wv.x, br0[g * BPG + p], a0);
      a1 = fmaf(wv.y, br1[g * BPG + p], a1);
    }
    rcv0 += 0.5f * tanhf(a0);
    rcv1 += 0.5f * tanhf(a1);
  }
#pragma unroll
  for (int c = 0; c < CMEM; ++c) {
    const float  iw = sigm(injw[n * CMEM + c]);
    const float2 v  = *(const float2*)(cc + ((b * TT + t) * CMEM + c) * DD + d0);
    rcv0 = fmaf(iw, v.x, rcv0);
    rcv1 = fmaf(iw, v.y, rcv1);
  }
  const float ed = eff_dec[b * NN + n];
  float2 hv = *(float2*)(hbuf + (b * NN + n) * DD + d0);
  const float h20 = fmaf(ed, hv.x - rcv0, rcv0);   // ed*h + (1-ed)*rcv
  const float h21 = fmaf(ed, hv.y - rcv1, rcv1);
  hv.x = h20; hv.y = h21;
  *(float2*)(hbuf + (b * NN + n) * DD + d0) = hv;
  const float2 ep = *(const float2*)(eff_prim + (b * NN + n) * DD + d0);
  float2 pmv; pmv.x = tanhf(h20 * ep.x); pmv.y = tanhf(h21 * ep.y);
  *(float2*)(pm_out + (b * NN + n) * DD + d0) = pmv;
}

// ---------------------------------------------------------------------------
// Readout: out[b,c,d] = sum_n sigmoid(ro[c,n]) * pm2[b,n,d] via fp32 WMMA.
// A (16x4): rows = c (8 real + 8 zero-padded), K = n chunk.
// B (4x16): cols = d tile of 16. Grid = B, block = 256 (8 waves split K).
// ---------------------------------------------------------------------------
__global__ __launch_bounds__(256) void k_readout(
    const float* __restrict__ pm, const float* __restrict__ rosig,
    float* __restrict__ ys, const int u)
{
  __shared__ float sred[8][4][8][32];   // 32 KB
  const int w = threadIdx.x >> 5;
  const int l = threadIdx.x & 31;
  const int b = blockIdx.x;

#if __has_builtin(__builtin_amdgcn_wmma_f32_16x16x4_f32)
  const int m  = l & 15;          // A row (c) / B column (d within tile)
  const int ka = (l >> 4) << 1;   // K pair: lanes 0-15 -> K0,K1; 16-31 -> K2,K3
  v8f acc[4] = {};
  const int n0 = w * 512;
  const float* arow = rosig + m * NN + ka;  // rows 8..15 are zero padding
  for (int it = 0; it < 128; ++it) {
    const int nb = n0 + it * 4;
    const float2 rv = *(const float2*)(arow + nb);
    v2f a; a.x = rv.x; a.y = rv.y;
    const float* p0 = pm + (b * NN + nb + ka) * DD + m;
    const float* p1 = p0 + DD;
#pragma unroll
    for (int t = 0; t < 4; ++t) {
      v2f bv; bv.x = p0[t * 16]; bv.y = p1[t * 16];
      acc[t] = __builtin_amdgcn_wmma_f32_16x16x4_f32(
          false, a, false, bv, 0, acc[t], false, false);
    }
  }
#pragma unroll
  for (int t = 0; t < 4; ++t)
#pragma unroll
    for (int r = 0; r < 8; ++r)
      sred[w][t][r][l] = acc[t][r];
#else
  // VALU fallback (same sred layout: sred[w][d>>4][c][d&15])
  const int n0 = w * 512;
  float accA[8], accB[8];
  for (int c = 0; c < 8; ++c) { accA[c] = 0.f; accB[c] = 0.f; }
  for (int i = 0; i < 512; ++i) {
    const int n = n0 + i;
    const float vA = pm[(b * NN + n) * DD + l];
    const float vB = pm[(b * NN + n) * DD + l + 32];
#pragma unroll
    for (int c = 0; c < 8; ++c) {
      const float r = rosig[c * NN + n];
      accA[c] = fmaf(r, vA, accA[c]);
      accB[c] = fmaf(r, vB, accB[c]);
    }
  }
  for (int c = 0; c < 8; ++c) {
    sred[w][l >> 4][c][l & 15] = accA[c];
    sred[w][(l + 32) >> 4][c][(l + 32) & 15] = accB[c];
  }
#endif
  __syncthreads();

  // Reduce 8 waves; D element (c,d): tile=d>>4, VGPR row c, lanes 0-15 = N.
  for (int j = threadIdx.x; j < CMEM * DD; j += 256) {
    const int c = j / DD;
    const int d = j % DD;
    const int tile = d >> 4;
    const int ln = d & 15;
    float s = 0.f;
#pragma unroll
    for (int wv = 0; wv < 8; ++wv) s += sred[wv][tile][c][ln];
#pragma unroll
    for (int rep = 0; rep < 4; ++rep) {   // pm is constant for 4 timesteps
      const int t = u * 4 + rep;
      ys[((b * TT + t) * CMEM + c) * DD + d] = s;
    }
  }
}

// ---------------------------------------------------------------------------
extern "C" void kernel_launch(void* const* d_in, const int* in_sizes, int n_in,
                              void* d_out, int out_size, void* d_ws, size_t ws_size,
                              hipStream_t stream)
{
  const float* cc   = (const float*)d_in[0];
  const float* hin  = (const float*)d_in[1];
  const float* pmin = (const float*)d_in[2];
  const float* tp   = (const float*)d_in[3];
  const float* tk   = (const float*)d_in[4];
  const float* prim = (const float*)d_in[5];
  const float* keyp = (const float*)d_in[6];
  const float* dlog = (const float*)d_in[7];
  const float* dbw  = (const float*)d_in[8];
  const float* dgw  = (const float*)d_in[9];
  const float* fc1w = (const float*)d_in[10];
  const float* fc1b = (const float*)d_in[11];
  const float* fc2w = (const float*)d_in[12];
  const float* fc2b = (const float*)d_in[13];
  const float* mlr  = (const float*)d_in[14];
  const float* injw = (const float*)d_in[15];
  const float* row  = (const float*)d_in[16];
  const int*   conn = (const int*)d_in[17];
  float* ys = (float*)d_out;

  float* ws = (float*)d_ws;
  const size_t SZ = (size_t)BB * NN * DD;   // 1,048,576 floats
  float* eff_prim = ws;
  float* eff_key  = ws + SZ;
  float* hbuf     = ws + 2 * SZ;
  float* pm0      = ws + 3 * SZ;
  float* pm1      = ws + 4 * SZ;
  float* eff_dec  = ws + 5 * SZ;
  float* rosig    = ws + 5 * SZ + (size_t)BB * NN;   // 16*NN floats (padded)

  k_init<<<dim3((BB * NN * DD) / 256), dim3(256), 0, stream>>>(
      hin, pmin, row, hbuf, pm0, rosig);
  k_modulator<<<dim3(NN / 8), dim3(256), 0, stream>>>(
      hin, tp, tk, prim, keyp, dlog, fc1w, fc1b, fc2w, fc2b, mlr,
      eff_prim, eff_key, eff_dec);

  float* pa = pm0;
  float* pb = pm1;
  for (int u = 0; u < NUPD; ++u) {
    const int t = u * 4;   // STRIDE
    k_update<<<dim3(NN * 2), dim3(64), 0, stream>>>(
        pa, pb, hbuf, eff_prim, eff_key, eff_dec, dbw, dgw, injw, cc, conn, t);
    k_readout<<<dim3(BB), dim3(256), 0, stream>>>(pb, rosig, ys, u);
    float* tmp = pa; pa = pb; pb = tmp;
  }
}